// DagEncoder_12721693131092
// MI455X (gfx1250) — compile-verified
//
#include <hip/hip_runtime.h>

// ---------------- problem constants (match reference) ----------------
#define N_NODES   4194304
#define NUM_SEGS  65536
#define DIM_EMBED 16
#define NFEAT     5

// ---------------- tiling / launch geometry ----------------
#define TILE        16                     // nodes per wave-tile (WMMA M)
#define TILES       (N_NODES / TILE)       // 262144
#define THREADS     256
#define WAVES_PB    (THREADS / 32)         // 8 waves (wave32)
#define BLOCKS      2048
#define TOTAL_WAVES (BLOCKS * WAVES_PB)    // 16384
#define TPW         (TILES / TOTAL_WAVES)  // 16 tiles per wave, exact

// LDS strides padded to avoid 64-bank conflicts
#define XSTR 25                            // 16 x 24 tile (K padded to 24)
#define HSTR 17                            // 16 x 16 tile
#define WAVE_LDS_FLOATS (16 * XSTR + 16 * HSTR)   // 672 floats / wave

typedef float v2f __attribute__((ext_vector_type(2)));
typedef float v8f __attribute__((ext_vector_type(8)));

__global__ void zero_out_kernel(float* __restrict__ out, int n) {
    for (int i = blockIdx.x * blockDim.x + threadIdx.x; i < n;
         i += gridDim.x * blockDim.x)
        out[i] = 0.0f;
}

__global__ __launch_bounds__(THREADS) void dag_encoder_kernel(
    const float* __restrict__ xfeat,   // [N,5]
    const float* __restrict__ emb,     // [N,16]
    const int*   __restrict__ ptr,     // [NUM_SEGS+1] CSR offsets
    const float* __restrict__ W1, const float* __restrict__ b1,   // [21,16],[16]
    const float* __restrict__ W2, const float* __restrict__ b2,   // [16,8],[8]
    const float* __restrict__ W3, const float* __restrict__ b3,   // [8,16],[16]
    float* __restrict__ out)           // [NUM_SEGS,16], pre-zeroed
{
    __shared__ float lds[WAVES_PB * WAVE_LDS_FLOATS];

    const int lane  = threadIdx.x & 31;
    const int wave  = threadIdx.x >> 5;
    const int gwave = blockIdx.x * WAVES_PB + wave;
    const int half  = lane >> 4;       // 0: lanes 0-15, 1: lanes 16-31
    const int r     = lane & 15;       // row (A/C) or column (B/C) index

    float* X = &lds[wave * WAVE_LDS_FLOATS];   // 16 x 24 input tile (stride 25)
    float* H = X + 16 * XSTR;                  // 16 x 16 hidden tile (stride 17)

    // ---- preload weight B-fragments (V_WMMA_F32_16X16X4_F32 layout) ----
    // B VGPR j, lane: K = kbase + 2*half + j, N = lane&15
    v2f wb1[6], wb2[4], wb3[2];
#pragma unroll
    for (int s = 0; s < 6; ++s) {              // K padded 21 -> 24
        const int k0 = 4 * s + 2 * half;
        wb1[s].x = (k0 + 0 < 21) ? W1[(k0 + 0) * 16 + r] : 0.0f;
        wb1[s].y = (k0 + 1 < 21) ? W1[(k0 + 1) * 16 + r] : 0.0f;
    }
#pragma unroll
    for (int s = 0; s < 4; ++s) {              // K = 16, N padded 8 -> 16
        const int k0 = 4 * s + 2 * half;
        wb2[s].x = (r < 8) ? W2[(k0 + 0) * 8 + r] : 0.0f;
        wb2[s].y = (r < 8) ? W2[(k0 + 1) * 8 + r] : 0.0f;
    }
#pragma unroll
    for (int s = 0; s < 2; ++s) {              // K = 8
        const int k0 = 4 * s + 2 * half;
        wb3[s].x = W3[(k0 + 0) * 16 + r];
        wb3[s].y = W3[(k0 + 1) * 16 + r];
    }
    const float cb1 = b1[r];
    const float cb2 = (r < 8) ? b2[r] : 0.0f;
    const float cb3 = b3[r];

#pragma unroll 1
    for (int t = 0; t < TPW; ++t) {
        const int tile = gwave * TPW + t;
        const int base = tile * TILE;
        const int node = base + r;             // this lane's staging row

        // ---- stage concat(x_feat, emb) into LDS, two lanes per row ----
        {
            const float4* ev = (const float4*)(emb + (size_t)node * DIM_EMBED);
            float* Xr = X + r * XSTR;
            if (half == 0) {
                const float4 e0 = ev[0], e1 = ev[1];
                const float* xf = xfeat + (size_t)node * NFEAT;
                Xr[0] = xf[0]; Xr[1] = xf[1]; Xr[2] = xf[2];
                Xr[3] = xf[3]; Xr[4] = xf[4];
                Xr[5]  = e0.x; Xr[6]  = e0.y; Xr[7]  = e0.z; Xr[8]  = e0.w;
                Xr[9]  = e1.x; Xr[10] = e1.y; Xr[11] = e1.z; Xr[12] = e1.w;
            } else {
                const float4 e2 = ev[2], e3 = ev[3];
                Xr[13] = e2.x; Xr[14] = e2.y; Xr[15] = e2.z; Xr[16] = e2.w;
                Xr[17] = e3.x; Xr[18] = e3.y; Xr[19] = e3.z; Xr[20] = e3.w;
                Xr[21] = 0.0f; Xr[22] = 0.0f; Xr[23] = 0.0f;   // K padding
            }
        }

        // ---- CSR segment id for this lane's row: searchsorted(ptr,node,'right')-1
        int lo = 0, hi = NUM_SEGS - 1;
        while (lo < hi) {
            const int mid = (lo + hi + 1) >> 1;
            if (ptr[mid] <= node) lo = mid; else hi = mid - 1;
        }
        const int seg = lo;

        asm volatile("s_wait_dscnt 0" ::: "memory");

        // ================= layer 1: X[16x24] @ W1 -> H1[16x16], ReLU =====
        v8f acc;
#pragma unroll
        for (int i = 0; i < 8; ++i) acc[i] = cb1;  // bias folded into C
        v2f a1[6];
#pragma unroll
        for (int s = 0; s < 6; ++s) {              // A layout: M=r, K=4s+2*half+{0,1}
            const float* p = X + r * XSTR + 4 * s + 2 * half;
            a1[s].x = p[0]; a1[s].y = p[1];
        }
#pragma unroll
        for (int s = 0; s < 6; ++s)
            acc = __builtin_amdgcn_wmma_f32_16x16x4_f32(
                false, a1[s], false, wb1[s], (short)0, acc, false, false);
#pragma unroll
        for (int i = 0; i < 8; ++i) acc[i] = fmaxf(acc[i], 0.0f);
        // C/D layout: VGPR i -> (M = i + 8*half, N = r)
#pragma unroll
        for (int i = 0; i < 8; ++i) H[(i + 8 * half) * HSTR + r] = acc[i];
        asm volatile("s_wait_dscnt 0" ::: "memory");

        // ================= layer 2: H1 @ W2 -> H2[16x8 (pad 16)], ReLU ===
        v2f a2[4];
#pragma unroll
        for (int s = 0; s < 4; ++s) {
            const float* p = H + r * HSTR + 4 * s + 2 * half;
            a2[s].x = p[0]; a2[s].y = p[1];
        }
#pragma unroll
        for (int i = 0; i < 8; ++i) acc[i] = cb2;
#pragma unroll
        for (int s = 0; s < 4; ++s)
            acc = __builtin_amdgcn_wmma_f32_16x16x4_f32(
                false, a2[s], false, wb2[s], (short)0, acc, false, false);
#pragma unroll
        for (int i = 0; i < 8; ++i) acc[i] = fmaxf(acc[i], 0.0f);
#pragma unroll
        for (int i = 0; i < 8; ++i) H[(i + 8 * half) * HSTR + r] = acc[i];
        asm volatile("s_wait_dscnt 0" ::: "memory");

        // ================= layer 3: H2 @ W3 -> D[16x16] (no activation) ==
        v2f a3[2];
#pragma unroll
        for (int s = 0; s < 2; ++s) {
            const float* p = H + r * HSTR + 4 * s + 2 * half;
            a3[s].x = p[0]; a3[s].y = p[1];
        }
#pragma unroll
        for (int i = 0; i < 8; ++i) acc[i] = cb3;
#pragma unroll
        for (int s = 0; s < 2; ++s)
            acc = __builtin_amdgcn_wmma_f32_16x16x4_f32(
                false, a3[s], false, wb3[s], (short)0, acc, false, false);
#pragma unroll
        for (int i = 0; i < 8; ++i) H[(i + 8 * half) * HSTR + r] = acc[i];
        asm volatile("s_wait_dscnt 0" ::: "memory");

        // ====== segment flush: run-length sum down each column, atomic per
        // segment boundary (rows are sorted by segment; ~1-2 segs/tile).
        // lanes 0-15: rows 0-7 of col r; lanes 16-31: rows 8-15 of col r.
        {
            const int mbase = 8 * half;
            int   cur  = __shfl(seg, mbase, 32);   // lane m holds seg of row m
            float sacc = 0.0f;
#pragma unroll
            for (int j = 0; j < 8; ++j) {
                const int   sj = __shfl(seg, mbase + j, 32);
                const float v  = H[(mbase + j) * HSTR + r];
                if (sj != cur) {
                    atomicAdd(&out[(size_t)cur * DIM_EMBED + r], sacc);
                    sacc = 0.0f;
                    cur  = sj;
                }
                sacc += v;
            }
            atomicAdd(&out[(size_t)cur * DIM_EMBED + r], sacc);
        }
    }
}

extern "C" void kernel_launch(void* const* d_in, const int* in_sizes, int n_in,
                              void* d_out, int out_size, void* d_ws, size_t ws_size,
                              hipStream_t stream) {
    const float* xfeat = (const float*)d_in[0];
    const float* emb   = (const float*)d_in[1];
    const int*   ptr   = (const int*)d_in[2];   // CSR offsets (harness int)
    const float* W1    = (const float*)d_in[3];
    const float* b1    = (const float*)d_in[4];
    const float* W2    = (const float*)d_in[5];
    const float* b2    = (const float*)d_in[6];
    const float* W3    = (const float*)d_in[7];
    const float* b3    = (const float*)d_in[8];
    float* out = (float*)d_out;

    // atomic accumulation target must start at zero every call
    zero_out_kernel<<<512, 256, 0, stream>>>(out, NUM_SEGS * DIM_EMBED);
    dag_encoder_kernel<<<BLOCKS, THREADS, 0, stream>>>(
        xfeat, emb, ptr, W1, b1, W2, b2, W3, b3, out);
}